// TAGCN_68135361184096
// MI455X (gfx1250) — compile-verified
//
#include <hip/hip_runtime.h>
#include <hip/hip_bf16.h>

// Problem constants (from reference)
#define NN     50000
#define EE     500000
#define FF     128
#define UNITS  256
#define KHOP   3
#define HCOLS  (FF * (KHOP + 1))   // 512

typedef __attribute__((ext_vector_type(2))) float v2f;
typedef __attribute__((ext_vector_type(8))) float v8f;

// Fire-and-forget fp32 atomic add executed at the L2 atomic units.
// Inline asm guarantees GLOBAL_ATOMIC_ADD_F32 (no CAS-loop expansion).
// No-return form -> tracked by STOREcnt, drained at kernel end; DEV scope
// so the RMW is performed at the device-coherent point (L2).
__device__ __forceinline__ void atom_add_f32(float* p, float v) {
    asm volatile("global_atomic_add_f32 %0, %1, off scope:SCOPE_DEV"
                 :: "v"(p), "v"(v) : "memory");
}

// ---------------- graph normalization ----------------

__global__ void zero_f32_kernel(float* __restrict__ p, int n) {
    int i = blockIdx.x * blockDim.x + threadIdx.x;
    if (i < n) p[i] = 0.0f;
}

__global__ void degree_kernel(const int* __restrict__ row,
                              const float* __restrict__ ew,
                              float* __restrict__ deg, int nE) {
    int e = blockIdx.x * blockDim.x + threadIdx.x;
    if (e < nE) atom_add_f32(&deg[row[e]], ew[e]);
}

__global__ void dinv_kernel(float* __restrict__ deg, int n) {
    int i = blockIdx.x * blockDim.x + threadIdx.x;
    if (i < n) {
        float d = deg[i];
        deg[i] = (d > 0.0f) ? rsqrtf(fmaxf(d, 1e-12f)) : 0.0f;
    }
}

__global__ void edgew_kernel(const int* __restrict__ row,
                             const int* __restrict__ col,
                             const float* __restrict__ ew,
                             const float* __restrict__ dinv,
                             float* __restrict__ w, int nE) {
    int e = blockIdx.x * blockDim.x + threadIdx.x;
    if (e < nE) w[e] = dinv[row[e]] * ew[e] * dinv[col[e]];
}

// ---------------- feature copies (x -> H, self-loop init) ----------------
// one wave of 32 lanes per node row; each lane moves a float4 (32*4 = 128 floats)

__global__ void copy_row128_kernel(float* __restrict__ dst, int dstStride,
                                   const float* __restrict__ src, int srcStride,
                                   int nNodes) {
    int gid  = blockIdx.x * blockDim.x + threadIdx.x;
    int node = gid >> 5;
    int lane = gid & 31;
    if (node >= nNodes) return;
    const float4* s = (const float4*)(src + (size_t)node * srcStride + lane * 4);
    float4* d       = (float4*)(dst + (size_t)node * dstStride + lane * 4);
    *d = *s;
}

// ---------------- edge scatter (one wave per edge) ----------------
// Gather 512B of features (float4/lane), scale by edge weight, scatter with
// 4 hardware fp32 atomics per lane. H is L2-resident (102 MB < 192 MB L2).

__global__ void scatter_kernel(const int* __restrict__ row,
                               const int* __restrict__ col,
                               const float* __restrict__ w,
                               float* __restrict__ H,
                               int prevCol, int nextCol, int nE) {
    int lane = threadIdx.x & 31;
    int e    = blockIdx.x * (blockDim.x >> 5) + (threadIdx.x >> 5);
    if (e >= nE) return;
    int   r  = row[e];
    int   c  = col[e];
    float ww = w[e];
    const float4* src = (const float4*)(H + (size_t)r * HCOLS + prevCol + lane * 4);
    float4 v = *src;
    float* dst = H + (size_t)c * HCOLS + nextCol + lane * 4;
    atom_add_f32(dst + 0, v.x * ww);
    atom_add_f32(dst + 1, v.y * ww);
    atom_add_f32(dst + 2, v.z * ww);
    atom_add_f32(dst + 3, v.w * ww);
}

// ---------------- dense tail GEMM: [N,512] @ [512,256] + bias ----------------
// fp32 WMMA 16x16x4. One wave computes one 16x16 C tile.
// 8 waves / 256-thread block; C has 50000 tiles; the launch gives 100000 blocks = 800000 waves (no tails,
// so EXEC stays all-ones as WMMA requires).
//
// Per ISA 7.12.2 layouts:
//   A (16x4, 2 VGPRs): lanes 0-15 hold M=lane, K={kb,kb+1}; lanes 16-31 K={kb+2,kb+3}
//   B (4x16, 2 VGPRs): lanes 0-15 hold N=lane, K={kb,kb+1}; lanes 16-31 K={kb+2,kb+3}
//   C (16x16, 8 VGPRs): VGPR v -> row v (lanes 0-15) / row v+8 (lanes 16-31), N=lane%16

__global__ __launch_bounds__(256) void gemm_wmma_kernel(
        const float* __restrict__ H,
        const float* __restrict__ Kmat,
        const float* __restrict__ bias,
        float* __restrict__ out) {
    int wave  = threadIdx.x >> 5;
    int lane  = threadIdx.x & 31;
    int tile  = blockIdx.x * 8 + wave;
    int mTile = tile >> 4;          // 0..49999
    int nTile = tile & 15;          // 0..15
    int mlane = lane & 15;
    int khalf = lane >> 4;          // 0 or 1

    int rowA = mTile * 16 + mlane;  // up to 799999 (no bound check)
    int colB = nTile * 16 + mlane;  // < 256

    const float* aPtr = H    + (size_t)rowA * HCOLS + khalf * 2;
    const float* bPtr = Kmat + (size_t)(khalf * 2) * UNITS + colB;

    v8f c = {};
#pragma unroll 4
    for (int kb = 0; kb < HCOLS; kb += 4) {
        v2f a = *(const v2f*)(aPtr + kb);
        v2f b;
        b.x = bPtr[(size_t)kb * UNITS];
        b.y = bPtr[(size_t)kb * UNITS + UNITS];
        c = __builtin_amdgcn_wmma_f32_16x16x4_f32(
                /*neg_a=*/false, a, /*neg_b=*/false, b,
                /*c_mod=*/(short)0, c, /*reuse_a=*/false, /*reuse_b=*/false);
    }

    float bv = bias[colB];
    int rbase = mTile * 16 + khalf * 8;
#pragma unroll
    for (int v = 0; v < 8; ++v) {
        out[(size_t)(rbase + v) * UNITS + colB] = c[v] + bv;
    }
}

// ---------------- host launcher ----------------

extern "C" void kernel_launch(void* const* d_in, const int* in_sizes, int n_in,
                              void* d_out, int out_size, void* d_ws, size_t ws_size,
                              hipStream_t stream) {
    const float* x    = (const float*)d_in[0];   // [N, F]
    const int*   eidx = (const int*)d_in[1];     // [2, E] flat: row | col
    const float* ew   = (const float*)d_in[2];   // [E]
    const float* kern = (const float*)d_in[3];   // [512, 256]
    const float* bias = (const float*)d_in[4];   // [256]
    float*       out  = (float*)d_out;           // [N, 256]

    const int* row = eidx;
    const int* col = eidx + EE;

    // workspace layout
    size_t degOff = 0;
    size_t wOff   = (degOff + (size_t)NN * 4 + 255) & ~(size_t)255;
    size_t hOff   = (wOff   + (size_t)EE * 4 + 255) & ~(size_t)255;
    float* deg = (float*)((char*)d_ws + degOff);
    float* w   = (float*)((char*)d_ws + wOff);
    float* H   = (float*)((char*)d_ws + hOff);   // [N, 512]

    const int BT = 256;

    // 1) deg = segment_sum(ew, row)
    zero_f32_kernel<<<(NN + BT - 1) / BT, BT, 0, stream>>>(deg, NN);
    degree_kernel<<<(EE + BT - 1) / BT, BT, 0, stream>>>(row, ew, deg, EE);
    // 2) deg -> deg_inv_sqrt (in place)
    dinv_kernel<<<(NN + BT - 1) / BT, BT, 0, stream>>>(deg, NN);
    // 3) w[e] = dinv[row]*ew*dinv[col]
    edgew_kernel<<<(EE + BT - 1) / BT, BT, 0, stream>>>(row, col, ew, deg, w, EE);

    // 4) H[:, 0:128] = x
    int copyThreads = NN * 32;
    copy_row128_kernel<<<(copyThreads + BT - 1) / BT, BT, 0, stream>>>(
        H, HCOLS, x, FF, NN);

    // 5) K propagation hops: next = prev (self-loop, weight 1) + scatter(edges)
    for (int k = 1; k <= KHOP; ++k) {
        int prevCol = (k - 1) * FF;
        int nextCol = k * FF;
        copy_row128_kernel<<<(copyThreads + BT - 1) / BT, BT, 0, stream>>>(
            H + nextCol, HCOLS, H + prevCol, HCOLS, NN);
        // one wave per edge, 8 waves per block
        scatter_kernel<<<(EE + 7) / 8, BT, 0, stream>>>(
            row, col, w, H, prevCol, nextCol, EE);
    }

    // 6) out = H @ kernel + bias   (fp32 WMMA, 100000 blocks, 8 waves/block)
    gemm_wmma_kernel<<<(NN * 16) / 8, BT, 0, stream>>>(H, kern, bias, out);
}